// SimpleAttention_8770323218849
// MI455X (gfx1250) — compile-verified
//
#include <hip/hip_runtime.h>
#include <hip/hip_bf16.h>

// ---------------- problem constants (match reference) ----------------
#define B_ROWS 16384
#define IN_DIM 128
#define H_DIM  128
#define NCLS   10
#define NATT   3
#define EPSLN  1e-5f
#define LOG2E  1.4426950408889634f

// ---------------- tiling ----------------
#define NWAVES 2            // 64 threads / WG (waves are fully independent)
#define RROWS  (NWAVES*16)  // 32 rows per WG (16 per wave)
#define LDH    132          // padded LDS row stride for h: bank=(4m+k)%64 -> conflict-free

typedef float v2f __attribute__((ext_vector_type(2)));
typedef float v8f __attribute__((ext_vector_type(8)));

// V_WMMA_F32_16X16X4_F32 : D(16x16 f32) = A(16x4 f32) * B(4x16 f32) + C
static __device__ __forceinline__ v8f wmma4(v2f a, v2f b, v8f c) {
  return __builtin_amdgcn_wmma_f32_16x16x4_f32(false, a, false, b, (short)0, c,
                                               false, false);
}

static __device__ __forceinline__ float fast_exp2(float x) {
#if __has_builtin(__builtin_amdgcn_exp2f)
  return __builtin_amdgcn_exp2f(x);      // v_exp_f32 (TRANS pipe)
#else
  return exp2f(x);
#endif
}
static __device__ __forceinline__ float fast_rcp(float x) {
#if __has_builtin(__builtin_amdgcn_rcpf)
  return __builtin_amdgcn_rcpf(x);       // v_rcp_f32
#else
  return 1.0f / x;
#endif
}
static __device__ __forceinline__ float fast_rsq(float x) {
#if __has_builtin(__builtin_amdgcn_rsqf)
  return __builtin_amdgcn_rsqf(x);       // v_rsq_f32
#else
  return rsqrtf(x);
#endif
}

__global__ void __launch_bounds__(NWAVES * 32, 1)
simpleattention_fused_kernel(const float* __restrict__ x,
                             const float* __restrict__ W_in,
                             const float* __restrict__ b_in,
                             const float* __restrict__ W_att,
                             const float* __restrict__ b_att,
                             const float* __restrict__ gamma,
                             const float* __restrict__ beta,
                             const float* __restrict__ W_c,
                             const float* __restrict__ b_c,
                             float* __restrict__ out) {
  extern __shared__ float smem[];
  float* sh_h = smem;                      // RROWS x LDH   (padded, 16B-aligned rows)
  float* sh_u = smem + RROWS * LDH;        // RROWS x H_DIM

  const int lane  = threadIdx.x & 31;
  const int wave  = threadIdx.x >> 5;
  const int mlan  = lane & 15;             // A row / B col / D col within tile
  const int half  = lane >> 4;             // half-wave id
  const int khalf = half << 1;             // K base offset {0,2} for A/B frags
  const int row0  = blockIdx.x * RROWS;    // global row base of this WG
  const int wrow  = wave * 16;             // this wave's 16-row tile inside WG

  // ---------------- stage 1: h = x @ W_in + b_in  (WMMA f32 16x16x4) -------
  {
    v8f acc[8] = {};
    const float* xrow = x + (size_t)(row0 + wrow + mlan) * IN_DIM;
    for (int kk = 0; kk < IN_DIM; kk += 4) {
      v2f a = *(const v2f*)(xrow + kk + khalf);   // A[m][k..k+1], 8B aligned
#pragma unroll
      for (int nt = 0; nt < 8; ++nt) {
        const int n = nt * 16 + mlan;
        v2f b;
        b.x = W_in[(kk + khalf) * H_DIM + n];
        b.y = W_in[(kk + khalf + 1) * H_DIM + n];
        acc[nt] = wmma4(a, b, acc[nt]);
      }
    }
#pragma unroll
    for (int nt = 0; nt < 8; ++nt) {
      const int n = nt * 16 + mlan;
      const float bias = b_in[n];
#pragma unroll
      for (int r = 0; r < 8; ++r) {
        const int m = wrow + r + 8 * half;        // D layout: M = r + 8*half
        sh_h[m * LDH + n] = acc[nt][r] + bias;
      }
    }
  }

  // ---------------- attention layers (wave-private; no barriers) -----------
  for (int layer = 0; layer < NATT; ++layer) {
    const float* W  = W_att + (size_t)layer * H_DIM * H_DIM;
    const float* bb = b_att + layer * H_DIM;

    // u = tanh(h @ W + b)
    {
      v8f acc[8] = {};
      const float* arow = sh_h + (wrow + mlan) * LDH;
      for (int kk = 0; kk < H_DIM; kk += 4) {
        v2f a = *(const v2f*)(arow + kk + khalf); // LDS, conflict-free
#pragma unroll
        for (int nt = 0; nt < 8; ++nt) {
          const int n = nt * 16 + mlan;
          v2f b;
          b.x = W[(kk + khalf) * H_DIM + n];
          b.y = W[(kk + khalf + 1) * H_DIM + n];
          acc[nt] = wmma4(a, b, acc[nt]);
        }
      }
#pragma unroll
      for (int nt = 0; nt < 8; ++nt) {
        const int n = nt * 16 + mlan;
        const float bias = bb[n];
#pragma unroll
        for (int r = 0; r < 8; ++r) {
          const int m = wrow + r + 8 * half;
          sh_u[m * H_DIM + n] = tanhf(acc[nt][r] + bias);
        }
      }
    }

    // rank-1 softmax attention + residual + layernorm, one row at a time.
    // out[i] = sum_j exp(u_i*u_j) h_j / sum_j exp(u_i*u_j); lane owns i = lane+32q.
    for (int rr = 0; rr < 16; ++rr) {
      const int row = wrow + rr;
      const float* ur = sh_u + row * H_DIM;
      float* hr = sh_h + row * LDH;

      float si[4], hi[4];
#pragma unroll
      for (int q = 0; q < 4; ++q) {
        const int i = lane + 32 * q;
        si[q] = ur[i] * LOG2E;                 // exp(u_i*u_j)=exp2(si*u_j)
        hi[q] = hr[i];
      }
      float Dv[4] = {0.f, 0.f, 0.f, 0.f};
      float Nv[4] = {0.f, 0.f, 0.f, 0.f};
      for (int j = 0; j < H_DIM; j += 4) {     // b128 LDS broadcast loads
        const float4 uj4 = *(const float4*)(ur + j);
        const float4 hj4 = *(const float4*)(hr + j);
        const float ujv[4] = {uj4.x, uj4.y, uj4.z, uj4.w};
        const float hjv[4] = {hj4.x, hj4.y, hj4.z, hj4.w};
#pragma unroll
        for (int c = 0; c < 4; ++c) {
#pragma unroll
          for (int q = 0; q < 4; ++q) {
            const float e = fast_exp2(si[q] * ujv[c]);   // v_exp_f32
            Dv[q] += e;
            Nv[q] += e * hjv[c];
          }
        }
      }
      float y[4], s = 0.f;
#pragma unroll
      for (int q = 0; q < 4; ++q) {
        y[q] = hi[q] + Nv[q] * fast_rcp(Dv[q]);    // residual
        s += y[q];
      }
#pragma unroll
      for (int off = 16; off > 0; off >>= 1) s += __shfl_xor(s, off, 32);
      const float mu = s * (1.0f / 128.0f);
      float v = 0.f;
#pragma unroll
      for (int q = 0; q < 4; ++q) {
        const float d = y[q] - mu;
        v += d * d;
      }
#pragma unroll
      for (int off = 16; off > 0; off >>= 1) v += __shfl_xor(v, off, 32);
      const float rstd = fast_rsq(v * (1.0f / 128.0f) + EPSLN);
#pragma unroll
      for (int q = 0; q < 4; ++q) {
        const int i = lane + 32 * q;
        hr[i] = (y[q] - mu) * rstd * gamma[layer * H_DIM + i] +
                beta[layer * H_DIM + i];
      }
    }
  }

  // ---------------- classifier: out = h @ W_c + b_c  (N padded 10->16) -----
  {
    v8f acc = {};
    const bool valid = (mlan < NCLS);
    const float* arow = sh_h + (wrow + mlan) * LDH;
    for (int kk = 0; kk < H_DIM; kk += 4) {
      v2f a = *(const v2f*)(arow + kk + khalf);
      v2f b;
      b.x = valid ? W_c[(kk + khalf) * NCLS + mlan] : 0.0f;
      b.y = valid ? W_c[(kk + khalf + 1) * NCLS + mlan] : 0.0f;
      acc = wmma4(a, b, acc);                // EXEC all-ones at the WMMA
    }
    if (valid) {
      const float bias = b_c[mlan];
#pragma unroll
      for (int r = 0; r < 8; ++r) {
        const int m = wrow + r + 8 * half;
        out[(size_t)(row0 + m) * NCLS + mlan] = acc[r] + bias;
      }
    }
  }
}

extern "C" void kernel_launch(void* const* d_in, const int* in_sizes, int n_in,
                              void* d_out, int out_size, void* d_ws,
                              size_t ws_size, hipStream_t stream) {
  const float* x     = (const float*)d_in[0];
  const float* W_in  = (const float*)d_in[1];
  const float* b_in  = (const float*)d_in[2];
  const float* W_att = (const float*)d_in[3];
  const float* b_att = (const float*)d_in[4];
  const float* gamma = (const float*)d_in[5];
  const float* beta  = (const float*)d_in[6];
  const float* W_c   = (const float*)d_in[7];
  const float* b_c   = (const float*)d_in[8];
  float* out = (float*)d_out;

  constexpr int kSmemBytes = (RROWS * LDH + RROWS * H_DIM) * (int)sizeof(float);
  // 33280 B dynamic LDS — under the 64 KB default cap, no attribute needed.

  dim3 grid(B_ROWS / RROWS);   // 512 workgroups
  dim3 block(NWAVES * 32);     // 64 threads = 2 wave32
  simpleattention_fused_kernel<<<grid, block, kSmemBytes, stream>>>(
      x, W_in, b_in, W_att, b_att, gamma, beta, W_c, b_c, out);
}